// HashGridEncoder_44495861186621
// MI455X (gfx1250) — compile-verified
//
#include <hip/hip_runtime.h>
#include <hip/hip_bf16.h>
#include <math.h>

// HashGridEncoder for MI455X (gfx1250).
//
// Roofline: 256 MB NT output stream (~11us @ 23.3 TB/s) + 2 GB of random
// 8-byte gathers into 64 MB of tables that fit the 192 MB L2. Gather-bound:
// keep 8 loads in flight per thread, 32M threads, tables L2-resident (RT
// loads), output stores non-temporal so the stream doesn't evict the tables.
// WMMA is deliberately NOT used: the 1x8 * 8x2 per-point reduction has no
// operand shared across lanes, so matrix ops would only add data-marshaling
// overhead to a 16-FMA payload. The gfx1250-specific paths used instead:
// global_load_async_to_lds_b32 (+ s_wait_asynccnt) and TH_NT stores.

typedef float v2f __attribute__((ext_vector_type(2)));

#define LEVELS 16
#define HASHMAP_SIZE 524288u
#define HASH_MASK (HASHMAP_SIZE - 1u)
#define THREADS_PER_POINT 16      // one thread per (point, level)
#define BLOCK_SIZE 256
#define POINTS_PER_BLOCK (BLOCK_SIZE / THREADS_PER_POINT)

#define AS1 __attribute__((address_space(1)))
#define AS3 __attribute__((address_space(3)))

struct ResTable { int res[LEVELS]; };

#if defined(__has_builtin)
#if __has_builtin(__builtin_amdgcn_global_load_async_to_lds_b32)
#define HAVE_ASYNC_LDS_BUILTIN 1
#endif
#endif

__global__ __launch_bounds__(BLOCK_SIZE) void
hashgrid_encode_kernel(const float* __restrict__ x,
                       const float* __restrict__ tables,
                       float* __restrict__ out,
                       int n_points, ResTable rt)
{
    __shared__ float sx[POINTS_PER_BLOCK * 3];

    const int tid      = threadIdx.x;
    const int block_p0 = blockIdx.x * POINTS_PER_BLOCK;

    // ---- Stage this block's 16 points (48 floats) into LDS via gfx1250
    // ---- async global->LDS copy (tracked by ASYNCcnt, no VGPR staging).
    {
        int nfloat = (n_points - block_p0) * 3;
        if (nfloat > POINTS_PER_BLOCK * 3) nfloat = POINTS_PER_BLOCK * 3;
        if (tid < nfloat) {
#if defined(HAVE_ASYNC_LDS_BUILTIN)
            __builtin_amdgcn_global_load_async_to_lds_b32(
                (AS1 int*)(AS1 void*)(x + (size_t)block_p0 * 3 + tid),
                (AS3 int*)(AS3 void*)(&sx[tid]),
                /*offset=*/0, /*cpol=*/0);
#else
            {
                unsigned loff = (unsigned)(size_t)(AS3 void*)((void*)&sx[tid]);
                const float* gsrc = x + (size_t)block_p0 * 3 + tid;
                asm volatile("global_load_async_to_lds_b32 %0, %1, off"
                             :: "v"(loff), "v"(gsrc) : "memory");
            }
#endif
        }
        asm volatile("s_wait_asynccnt 0" ::: "memory");
        __syncthreads();
    }

    const int p_local = tid >> 4;                       // local point in block
    const int l       = tid & (THREADS_PER_POINT - 1);  // level
    const int n       = block_p0 + p_local;
    if (n >= n_points) return;

    const float px = sx[p_local * 3 + 0];
    const float py = sx[p_local * 3 + 1];
    const float pz = sx[p_local * 3 + 2];

    // pos = x * res (f32, matches jnp); trilinear cell + fractional weights
    const float res = (float)rt.res[l];
    const float fx = px * res, fy = py * res, fz = pz * res;
    const float flx = floorf(fx), fly = floorf(fy), flz = floorf(fz);
    const float wx = fx - flx, wy = fy - fly, wz = fz - flz;
    const int bx = (int)flx, by = (int)fly, bz = (int)flz;

    // Spatial hash (uint32 wraparound mul, xor, power-of-two mask).
    const uint32_t hx0 = (uint32_t)bx       * 73856093u;
    const uint32_t hx1 = (uint32_t)(bx + 1) * 73856093u;
    const uint32_t hy0 = (uint32_t)by       * 19349663u;
    const uint32_t hy1 = (uint32_t)(by + 1) * 19349663u;
    const uint32_t hz0 = (uint32_t)bz       * 83492791u;
    const uint32_t hz1 = (uint32_t)(bz + 1) * 83492791u;

    uint32_t idx[8];
    idx[0] = (hx0 ^ hy0 ^ hz0) & HASH_MASK;
    idx[1] = (hx0 ^ hy0 ^ hz1) & HASH_MASK;
    idx[2] = (hx0 ^ hy1 ^ hz0) & HASH_MASK;
    idx[3] = (hx0 ^ hy1 ^ hz1) & HASH_MASK;
    idx[4] = (hx1 ^ hy0 ^ hz0) & HASH_MASK;
    idx[5] = (hx1 ^ hy0 ^ hz1) & HASH_MASK;
    idx[6] = (hx1 ^ hy1 ^ hz0) & HASH_MASK;
    idx[7] = (hx1 ^ hy1 ^ hz1) & HASH_MASK;

    // 8 independent 8-byte gathers -> all issued before first use so the
    // compiler keeps LOADcnt at 8 and waits once. Tables are L2-resident
    // (64 MB < 192 MB L2), default RT temporal hint keeps them there.
    const v2f* __restrict__ tl = (const v2f*)tables + (size_t)l * HASHMAP_SIZE;
    v2f v[8];
#pragma unroll
    for (int c = 0; c < 8; ++c) v[c] = tl[idx[c]];

    // Trilinear corner weights: prod over axes of (offset ? w : 1-w).
    const float w0x = 1.0f - wx, w0y = 1.0f - wy, w0z = 1.0f - wz;
    const float wyz00 = w0y * w0z, wyz01 = w0y * wz;
    const float wyz10 = wy  * w0z, wyz11 = wy  * wz;
    const float cw[8] = { w0x * wyz00, w0x * wyz01, w0x * wyz10, w0x * wyz11,
                          wx  * wyz00, wx  * wyz01, wx  * wyz10, wx  * wyz11 };

    v2f acc = {0.0f, 0.0f};
#pragma unroll
    for (int c = 0; c < 8; ++c) acc += v[c] * cw[c];

    // Coalesced (256 B contiguous per wave) non-temporal store: the 256 MB
    // output stream must not evict the L2-resident tables.
    v2f* o = (v2f*)out + (size_t)n * LEVELS + l;
    __builtin_nontemporal_store(acc, o);
}

extern "C" void kernel_launch(void* const* d_in, const int* in_sizes, int n_in,
                              void* d_out, int out_size, void* d_ws, size_t ws_size,
                              hipStream_t stream) {
    (void)n_in; (void)out_size; (void)d_ws; (void)ws_size;
    const float* x      = (const float*)d_in[0];
    const float* tables = (const float*)d_in[1];
    float*       out    = (float*)d_out;

    const int n_points = in_sizes[0] / 3;

    // Exactly match Python's int(16 * 1.3819**l) via double pow.
    ResTable rt;
    for (int l = 0; l < LEVELS; ++l)
        rt.res[l] = (int)(16.0 * pow(1.3819, (double)l));

    const int blocks = (n_points + POINTS_PER_BLOCK - 1) / POINTS_PER_BLOCK;
    hipLaunchKernelGGL(hashgrid_encode_kernel, dim3(blocks), dim3(BLOCK_SIZE), 0, stream,
                       x, tables, out, n_points, rt);
}